// AttentionFlowLayer_78847009620105
// MI455X (gfx1250) — compile-verified
//
#include <hip/hip_runtime.h>
#include <math.h>

// B=64, T=1024, J=64, d=256. out = concat([H, c2q, H*c2q, H*q2c], dim=2) -> (B,T,1024)
// Memory-bound (~450MB traffic @23.3TB/s ~= 19us); full-FP32 WMMA keeps reference
// numerics while compute (~4.3 GFLOP via v_wmma_f32_16x16x4_f32) stays far below the
// bandwidth roofline. GEMM2 A-fragments (softmax probs) are hoisted into registers
// and reused across all 8 N-tiles to cut LDS load pressure per WMMA.

typedef __attribute__((ext_vector_type(2))) float v2f;
typedef __attribute__((ext_vector_type(8))) float v8f;

#define T_DIM 1024
#define J_DIM 64
#define D_DIM 256
#define TM    64          // rows of T per workgroup in pass1

// LDS float offsets for pass1
#define OFF_H  0          // 64*256 = 16384
#define OFF_U  16384      // 64*256 = 16384
#define OFF_S  32768      // 64*64  = 4096
#define OFF_W  36864      // 768
#define OFF_HD 37632      // 64
#define OFF_UD 37696      // 64
#define LDS_FLOATS 37760  // ~151 KB -> 2 workgroups per WGP (302KB of 320KB LDS)

__global__ __launch_bounds__(256)
void bidaf_pass1(const float* __restrict__ Hg, const float* __restrict__ Ug,
                 const float* __restrict__ Wg, const float* __restrict__ bg,
                 float* __restrict__ out, float* __restrict__ wsM)
{
    extern __shared__ float sm[];
    float* Hs   = sm + OFF_H;
    float* Us   = sm + OFF_U;
    float* Ss   = sm + OFF_S;
    float* Ws   = sm + OFF_W;
    float* hdot = sm + OFF_HD;
    float* udot = sm + OFF_UD;

    const int tid = threadIdx.x;
    const int blk = blockIdx.x;
    const int b   = blk >> 4;            // 16 T-tiles per batch
    const int t0  = (blk & 15) * TM;

    const size_t hbase = ((size_t)b * T_DIM + t0) * D_DIM;
    const size_t ubase = (size_t)b * J_DIM * D_DIM;

    // Stage tiles (fully coalesced: tiles are contiguous in memory)
    for (int i = tid; i < TM * D_DIM; i += 256)     Hs[i] = Hg[hbase + i];
    for (int i = tid; i < J_DIM * D_DIM; i += 256)  Us[i] = Ug[ubase + i];
    for (int i = tid; i < 3 * D_DIM; i += 256)      Ws[i] = Wg[i];
    __syncthreads();

    // Per-row dot products: hdot[m] = H[m]·w_h ; udot[j] = U[j]·w_u
    if (tid < 64) {
        float s = 0.f;
        for (int k = 0; k < D_DIM; ++k) s += Hs[tid * D_DIM + k] * Ws[k];
        hdot[tid] = s;
    } else if (tid < 128) {
        const int j = tid - 64;
        float s = 0.f;
        for (int k = 0; k < D_DIM; ++k) s += Us[j * D_DIM + k] * Ws[D_DIM + k];
        udot[j] = s;
    }
    __syncthreads();

    const float bias = bg[0];
    const int wave = tid >> 5;
    const int lane = tid & 31;
    const int half = lane >> 4;   // K/M-half selector per ISA layouts
    const int ln   = lane & 15;   // N (B/C) or M (A) within half

    // ---------------- GEMM1: S = (H .* w_hu) @ U^T  (64x64, K=256) --------------
    {
        const int m0 = (wave & 3) * 16;
        const int j0 = (wave >> 2) * 32;          // this wave does j0 and j0+16
        v8f c0 = {}; v8f c1 = {};
        for (int k0 = 0; k0 < D_DIM; k0 += 4) {
            const int ka = k0 + 2 * half;
            v2f a;                                 // A 16x4: lane=M, {v0,v1}={K=2h,2h+1}
            a.x = Hs[(m0 + ln) * D_DIM + ka]     * Ws[2 * D_DIM + ka];
            a.y = Hs[(m0 + ln) * D_DIM + ka + 1] * Ws[2 * D_DIM + ka + 1];
            v2f b0, b1;                            // B 4x16: lane=N, {v0,v1}={K=2h,2h+1}; B[k][j]=U[j][k]
            b0.x = Us[(j0 + ln) * D_DIM + ka];
            b0.y = Us[(j0 + ln) * D_DIM + ka + 1];
            b1.x = Us[(j0 + 16 + ln) * D_DIM + ka];
            b1.y = Us[(j0 + 16 + ln) * D_DIM + ka + 1];
            c0 = __builtin_amdgcn_wmma_f32_16x16x4_f32(false, a, false, b0, (short)0, c0, false, false);
            c1 = __builtin_amdgcn_wmma_f32_16x16x4_f32(false, a, false, b1, (short)0, c1, false, false);
        }
        // C 16x16: lane=N, m = r + 8*half. Add rank-1 terms + bias, stash S in LDS.
        for (int r = 0; r < 8; ++r) {
            const int m = m0 + r + 8 * half;
            Ss[m * J_DIM + j0 + ln]      = c0[r] + hdot[m] + udot[j0 + ln] + bias;
            Ss[m * J_DIM + j0 + 16 + ln] = c1[r] + hdot[m] + udot[j0 + 16 + ln] + bias;
        }
    }
    __syncthreads();

    // ---------------- row softmax over J=64 (also record row max) ----------------
    if (tid < TM) {
        float mx = -3.402823466e38f;
        for (int j = 0; j < J_DIM; ++j) mx = fmaxf(mx, Ss[tid * J_DIM + j]);
        wsM[(size_t)b * T_DIM + t0 + tid] = mx;        // for q2c softmax (pass2)
        float sum = 0.f;
        for (int j = 0; j < J_DIM; ++j) {
            const float e = __expf(Ss[tid * J_DIM + j] - mx);
            Ss[tid * J_DIM + j] = e;
            sum += e;
        }
        const float inv = 1.f / sum;
        for (int j = 0; j < J_DIM; ++j) Ss[tid * J_DIM + j] *= inv;
    }
    __syncthreads();

    // ---------------- GEMM2: c2q = P @ U  (64x256, K=64) -------------------------
    {
        const int m0 = (wave & 3) * 16;

        // Hoist all A-fragments (P rows, invariant across the 8 N-tiles) into
        // registers: 16 k-steps x v2f = 32 VGPRs, loaded from LDS exactly once.
        v2f afrag[16];
#pragma unroll
        for (int ks = 0; ks < 16; ++ks) {
            const int ka = ks * 4 + 2 * half;
            afrag[ks].x = Ss[(m0 + ln) * J_DIM + ka];
            afrag[ks].y = Ss[(m0 + ln) * J_DIM + ka + 1];
        }

        for (int nt = 0; nt < 8; ++nt) {
            const int n0 = (wave >> 2) * 128 + nt * 16;
            v8f c = {};
#pragma unroll
            for (int ks = 0; ks < 16; ++ks) {
                const int ka = ks * 4 + 2 * half;
                v2f bb;                              // B[k][n] = U[k][n]
                bb.x = Us[ka * D_DIM + n0 + ln];
                bb.y = Us[(ka + 1) * D_DIM + n0 + ln];
                c = __builtin_amdgcn_wmma_f32_16x16x4_f32(false, afrag[ks], false, bb, (short)0, c, false, false);
            }
#pragma unroll
            for (int r = 0; r < 8; ++r) {
                const int m = m0 + r + 8 * half;
                const size_t row = ((size_t)b * T_DIM + t0 + m) * (4 * D_DIM);
                const float cv = c[r];
                const float hv = Hs[m * D_DIM + n0 + ln];
                out[row + D_DIM + n0 + ln]     = cv;       // chunk1: c2q
                out[row + 2 * D_DIM + n0 + ln] = hv * cv;  // chunk2: H*c2q
            }
        }
    }
}

// Per-batch softmax over the T row-maxes (in place in wsM); zero q2c accumulator.
__global__ __launch_bounds__(256)
void bidaf_pass2a(float* __restrict__ wsM, float* __restrict__ q2c)
{
    __shared__ float red[256];
    const int b = blockIdx.x;
    const int tid = threadIdx.x;
    float* M = wsM + (size_t)b * T_DIM;
    float v[4];
    float mx = -3.402823466e38f;
    for (int i = 0; i < 4; ++i) { v[i] = M[tid * 4 + i]; mx = fmaxf(mx, v[i]); }
    red[tid] = mx; __syncthreads();
    for (int s = 128; s > 0; s >>= 1) { if (tid < s) red[tid] = fmaxf(red[tid], red[tid + s]); __syncthreads(); }
    mx = red[0]; __syncthreads();
    float e[4]; float sum = 0.f;
    for (int i = 0; i < 4; ++i) { e[i] = __expf(v[i] - mx); sum += e[i]; }
    red[tid] = sum; __syncthreads();
    for (int s = 128; s > 0; s >>= 1) { if (tid < s) red[tid] += red[tid + s]; __syncthreads(); }
    const float inv = 1.f / red[0];
    for (int i = 0; i < 4; ++i) M[tid * 4 + i] = e[i] * inv;   // p[b,t]
    q2c[(size_t)b * D_DIM + tid] = 0.f;                        // zero accumulator
}

// q2c[b,:] = sum_t p[b,t] * H[b,t,:]  (16 T-chunks per batch, atomic partials)
__global__ __launch_bounds__(256)
void bidaf_pass2b(const float* __restrict__ Hg, const float* __restrict__ p,
                  float* __restrict__ q2c)
{
    const int b  = blockIdx.x >> 4;
    const int t0 = (blockIdx.x & 15) * 64;
    const int tid = threadIdx.x;
    const float* Hb = Hg + ((size_t)b * T_DIM + t0) * D_DIM;
    const float* pb = p + (size_t)b * T_DIM + t0;
    float acc = 0.f;
    for (int r = 0; r < 64; ++r) acc += pb[r] * Hb[r * D_DIM + tid];
    atomicAdd(&q2c[(size_t)b * D_DIM + tid], acc);
}

// chunk0 = H, chunk3 = H * q2c (broadcast over t); one row per block, coalesced.
__global__ __launch_bounds__(256)
void bidaf_pass3(const float* __restrict__ Hg, const float* __restrict__ q2c,
                 float* __restrict__ out)
{
    const size_t row = blockIdx.x;        // b*T + t
    const int b = blockIdx.x >> 10;
    const int tid = threadIdx.x;
    const float h = Hg[row * D_DIM + tid];
    const float q = q2c[(size_t)b * D_DIM + tid];
    out[row * (4 * D_DIM) + tid] = h;
    out[row * (4 * D_DIM) + 3 * D_DIM + tid] = h * q;
}

extern "C" void kernel_launch(void* const* d_in, const int* in_sizes, int n_in,
                              void* d_out, int out_size, void* d_ws, size_t ws_size,
                              hipStream_t stream)
{
    const float* H  = (const float*)d_in[0];
    const float* U  = (const float*)d_in[1];
    const float* W  = (const float*)d_in[2];
    const float* bb = (const float*)d_in[3];
    float* out = (float*)d_out;

    float* wsM = (float*)d_ws;                 // B*T floats (row max -> softmax p)
    float* q2c = wsM + 64 * T_DIM;             // B*256 floats

    const size_t lds = LDS_FLOATS * sizeof(float);
    // Opt in to >64KB dynamic LDS (deterministic host-side call; capture-safe).
    (void)hipFuncSetAttribute((const void*)bidaf_pass1,
                              hipFuncAttributeMaxDynamicSharedMemorySize, (int)lds);

    bidaf_pass1<<<dim3(64 * 16), dim3(256), lds, stream>>>(H, U, W, bb, out, wsM);
    bidaf_pass2a<<<dim3(64), dim3(256), 0, stream>>>(wsM, q2c);
    bidaf_pass2b<<<dim3(64 * 16), dim3(256), 0, stream>>>(H, wsM, q2c);
    bidaf_pass3<<<dim3(64 * T_DIM), dim3(256), 0, stream>>>(H, q2c, out);
}